// TopKGating_9363028706162
// MI455X (gfx1250) — compile-verified
//
#include <hip/hip_runtime.h>
#include <hip/hip_bf16.h>
#include <math.h>

// TopKGating (eval path): gates = softmax( decomposition_tp( x @ W^T + b ) )
// x: [131072, 1024] f32, W: [8, 1024] f32, b: [8] f32, out: [131072, 8] f32.
// Memory bound: 512 MB of x @ 23.3 TB/s ~= 22 us. GEMM via V_WMMA_F32_16X16X4_F32.

#define DIM       1024
#define NEXP      8
#define ALPHA     10.0f
#define WROW      1026           // padded LDS row stride (floats) for gate_w
#define WAVES_PB  8
#define ROWS_PW   16

typedef float v2f __attribute__((ext_vector_type(2)));
typedef float v8f __attribute__((ext_vector_type(8)));

__global__ __launch_bounds__(256) void topk_gating_wmma(
    const float* __restrict__ x,
    const float* __restrict__ gate_w,
    const float* __restrict__ gate_b,
    float* __restrict__ out)
{
  __shared__ float sW[9 * WROW];                      // rows 0..7 experts, row 8 = zeros
  __shared__ float sB[NEXP];
  __shared__ float sC[WAVES_PB][ROWS_PW][17];         // per-wave C transpose tile

  const int tid  = threadIdx.x;
  const int wave = tid >> 5;
  const int lane = tid & 31;

  // ---- stage gate_w (+ zero row) and gate_b into LDS ----
  for (int i = tid; i < 9 * WROW; i += 256) {
    int r = i / WROW;
    int c = i - r * WROW;
    float v = 0.0f;
    if (r < NEXP && c < DIM) v = gate_w[r * DIM + c];
    sW[i] = v;
  }
  if (tid < NEXP) sB[tid] = gate_b[tid];
  __syncthreads();

  // ---- WMMA fragment addressing (A: 16x4 f32, B: 4x16 f32) ----
  // A layout: lanes 0-15 hold M=0..15; VGPR0 = K{0,2}, VGPR1 = K{1,3} -> per-lane
  // contiguous float2 at column offset 2*hi. B mirrors it with N = lane%16.
  const int m  = lane & 15;                           // A row / B column (expert)
  const int hi = lane >> 4;                           // K half select
  const int tile    = blockIdx.x * WAVES_PB + wave;   // one 16-row tile per wave
  const int rowBase = tile * ROWS_PW;

  const float* aPtr = x  + (size_t)(rowBase + m) * DIM + 2 * hi;
  const float* bPtr = sW + (m < NEXP ? m : NEXP) * WROW + 2 * hi;

  v8f acc = {};
  for (int ko = 0; ko < DIM; ko += 32) {
    __builtin_prefetch(aPtr + ko + 512, 0, 0);        // ~2KB ahead -> global_prefetch_b8
#pragma unroll
    for (int ki = 0; ki < 32; ki += 4) {
      v2f a = *(const v2f*)(aPtr + ko + ki);
      v2f b = *(const v2f*)(bPtr + ko + ki);
      acc = __builtin_amdgcn_wmma_f32_16x16x4_f32(false, a, false, b,
                                                  (short)0, acc, false, false);
    }
  }

  // ---- transpose C through LDS: acc[r] is (row = r + 8*hi, col = m) ----
#pragma unroll
  for (int r = 0; r < 8; ++r)
    sC[wave][r + 8 * hi][m] = acc[r];
  __syncthreads();

  // ---- per-row tail: lanes 0..15 each own one row of the tile ----
  if (lane < ROWS_PW) {
    float lg[NEXP];
    float m1 = -INFINITY, m2 = -INFINITY;             // largest / 2nd-largest
#pragma unroll
    for (int e = 0; e < NEXP; ++e) {
      float v = sC[wave][lane][e] + sB[e];
      lg[e] = v;
      if (v > m1) { m2 = m1; m1 = v; }
      else if (v > m2) { m2 = v; }
    }
    // s = softmax(logits) (stable)
    float se[NEXP], ssum = 0.0f;
#pragma unroll
    for (int e = 0; e < NEXP; ++e) { se[e] = expf(lg[e] - m1); ssum += se[e]; }
    float sinv = 1.0f / ssum;
    // out = mask ? a*log1p(s) : a*(exp(s)-1);  mask = logit < kth-largest
    float o[NEXP], omx = -INFINITY;
#pragma unroll
    for (int e = 0; e < NEXP; ++e) {
      float s = se[e] * sinv;
      float v = (lg[e] < m2) ? ALPHA * log1pf(s) : ALPHA * (expf(s) - 1.0f);
      o[e] = v;
      omx = fmaxf(omx, v);
    }
    // gates = softmax(out)
    float osum = 0.0f;
#pragma unroll
    for (int e = 0; e < NEXP; ++e) { o[e] = expf(o[e] - omx); osum += o[e]; }
    float oinv = 1.0f / osum;

    float4* dst = (float4*)(out + (size_t)(rowBase + lane) * NEXP);
    dst[0] = make_float4(o[0]*oinv, o[1]*oinv, o[2]*oinv, o[3]*oinv);
    dst[1] = make_float4(o[4]*oinv, o[5]*oinv, o[6]*oinv, o[7]*oinv);
  }
}

extern "C" void kernel_launch(void* const* d_in, const int* in_sizes, int n_in,
                              void* d_out, int out_size, void* d_ws, size_t ws_size,
                              hipStream_t stream) {
  const float* x      = (const float*)d_in[0];
  const float* gate_w = (const float*)d_in[1];
  const float* gate_b = (const float*)d_in[2];
  float* out          = (float*)d_out;

  const int rows = in_sizes[0] / DIM;                 // 131072
  const int rows_per_block = WAVES_PB * ROWS_PW;      // 128
  const int grid = rows / rows_per_block;             // 1024

  topk_gating_wmma<<<grid, 256, 0, stream>>>(x, gate_w, gate_b, out);
}